// TestAllReduceFusedAddRMSNormModel_12592844112269
// MI455X (gfx1250) — compile-verified
//
#include <hip/hip_runtime.h>

// AllReduce(sum over TP=4) + residual add + RMSNorm, fp32.
// T=8192 rows, H=4096. Memory-bound: ~768 MiB traffic -> ~34us floor @ 23.3 TB/s.
// Strategy: one 256-thread block (8 wave32s) per row; B128 non-temporal loads
// for the 640 MiB of streamed data (won't fit 192MB L2, don't pollute it);
// weight (16 KiB) loaded with default temporal hint so it stays cache-resident;
// row slice kept in registers between the reduce and normalize passes.

typedef __attribute__((ext_vector_type(4))) float v4f;

constexpr int   kT   = 8192;
constexpr int   kH   = 4096;
constexpr int   kTPB = 256;                    // 8 waves (wave32)
constexpr int   kV4PerRow    = kH / 4;         // 1024 float4 per row
constexpr int   kV4PerThread = kV4PerRow / kTPB; // 4 float4 per thread
constexpr float kEps = 1e-6f;

__global__ __launch_bounds__(kTPB)
void allreduce_add_rmsnorm_kernel(const float* __restrict__ hs,   // [4, T, H]
                                  const float* __restrict__ res,  // [T, H]
                                  const float* __restrict__ w,    // [H]
                                  float* __restrict__ out)        // [T, H]
{
    const int    row         = blockIdx.x;
    const int    tid         = threadIdx.x;
    const size_t rowBase     = (size_t)row * kH;
    const size_t sliceStride = (size_t)kT * kH;

    const v4f* __restrict__ h0 = (const v4f*)(hs + rowBase);
    const v4f* __restrict__ h1 = (const v4f*)(hs + rowBase + 1 * sliceStride);
    const v4f* __restrict__ h2 = (const v4f*)(hs + rowBase + 2 * sliceStride);
    const v4f* __restrict__ h3 = (const v4f*)(hs + rowBase + 3 * sliceStride);
    const v4f* __restrict__ rv = (const v4f*)(res + rowBase);
    const v4f* __restrict__ wv = (const v4f*)w;
    v4f*       __restrict__ ov = (v4f*)(out + rowBase);

    // Pass 1: stream 5 inputs (NT), accumulate x = sum + residual in registers,
    // and accumulate per-thread sum of squares.
    v4f   x[kV4PerThread];
    float ssq = 0.0f;

#pragma unroll
    for (int i = 0; i < kV4PerThread; ++i) {
        const int idx = tid + i * kTPB;            // block-contiguous, coalesced
        v4f a = __builtin_nontemporal_load(&h0[idx]);
        v4f b = __builtin_nontemporal_load(&h1[idx]);
        v4f c = __builtin_nontemporal_load(&h2[idx]);
        v4f d = __builtin_nontemporal_load(&h3[idx]);
        v4f r = __builtin_nontemporal_load(&rv[idx]);
        v4f s = a + b + c + d + r;
        x[i]  = s;
        ssq  += s.x * s.x + s.y * s.y + s.z * s.z + s.w * s.w;
    }

    // Wave32 reduction of ssq (lowers to LDS-permute hardware, no LDS traffic).
#pragma unroll
    for (int off = 16; off > 0; off >>= 1) {
        ssq += __shfl_xor(ssq, off, 32);
    }

    // Cross-wave reduction: 8 partials through LDS, broadcast back.
    __shared__ float warpSsq[kTPB / 32];
    const int lane = tid & 31;
    const int wid  = tid >> 5;
    if (lane == 0) warpSsq[wid] = ssq;
    __syncthreads();

    float total = 0.0f;
#pragma unroll
    for (int i = 0; i < kTPB / 32; ++i) total += warpSsq[i];

    const float inv = rsqrtf(total * (1.0f / (float)kH) + kEps); // v_rsq_f32

    // Pass 2: scale by rsqrt and weight; weight uses default (temporal) hint so
    // the 16 KiB vector stays hot in cache across all 8192 rows; output is
    // streamed out non-temporally.
#pragma unroll
    for (int i = 0; i < kV4PerThread; ++i) {
        const int idx = tid + i * kTPB;
        v4f ww = wv[idx];
        v4f o  = x[i] * inv * ww;
        __builtin_nontemporal_store(o, &ov[idx]);
    }
}

extern "C" void kernel_launch(void* const* d_in, const int* in_sizes, int n_in,
                              void* d_out, int out_size, void* d_ws, size_t ws_size,
                              hipStream_t stream)
{
    const float* hs  = (const float*)d_in[0];  // hidden_states [4, 8192, 4096]
    const float* res = (const float*)d_in[1];  // residual      [8192, 4096]
    const float* w   = (const float*)d_in[2];  // weight        [4096]
    float*       out = (float*)d_out;          // normed        [8192, 4096]

    dim3 grid(kT);
    dim3 block(kTPB);
    allreduce_add_rmsnorm_kernel<<<grid, block, 0, stream>>>(hs, res, w, out);
}